// MambaCompressor_19473381720425
// MI455X (gfx1250) — compile-verified
//
#include <hip/hip_runtime.h>
#include <hip/hip_bf16.h>

// Problem constants (match reference)
#define B_DIM 16
#define S_DIM 4096
#define H_DIM 1024
#define D_DIM 4096
#define M_DIM 64          // S / MEM_EVERY
#define LN_EPS 1e-5f

typedef __attribute__((ext_vector_type(2))) float v2f;
typedef __attribute__((ext_vector_type(8))) float v8f;

// ---------------------------------------------------------------------------
// Kernel 1: find positions of mem tokens. One wave32 per batch row.
// Sequential scan in 32-token chunks using ballot + prefix popcount keeps the
// ordering identical to jnp.cumsum(mask).
// ---------------------------------------------------------------------------
__global__ void find_mem_kernel(const int* __restrict__ ids,
                                const int* __restrict__ mem_id_ptr,
                                int* __restrict__ posIdx) {
    const int b    = blockIdx.x;
    const int lane = threadIdx.x;            // 0..31 (wave32)
    const int mem_id = mem_id_ptr[0];
    const int* row = ids + b * S_DIM;

    int base = 0;  // uniform across the wave
    for (int s0 = 0; s0 < S_DIM; s0 += 32) {
        const bool is_mem = (row[s0 + lane] == mem_id);
        const unsigned mask = (unsigned)__ballot(is_mem);   // wave32: low 32 bits
        const int prefix = __popc(mask & ((1u << lane) - 1u));
        if (is_mem) {
            const int slot = base + prefix;
            if (slot < M_DIM) posIdx[b * M_DIM + slot] = s0 + lane;
        }
        base += __popc(mask);
    }
    // Defensive: zero-fill unused slots (input guarantees exactly M per row).
    for (int i = base + lane; i < M_DIM; i += 32) posIdx[b * M_DIM + i] = 0;
}

// ---------------------------------------------------------------------------
// Kernel 2: gather selected rows + LayerNorm -> dense fp32 panel normed[1024][1024].
// One 256-thread block (8 waves) per selected row. Only the B*M selected rows
// of `hidden` (4 MB of 256 MB) are ever read.
// ---------------------------------------------------------------------------
__global__ void ln_gather_kernel(const float* __restrict__ hidden,
                                 const float* __restrict__ gamma,
                                 const float* __restrict__ beta,
                                 const int* __restrict__ posIdx,
                                 float* __restrict__ normed) {
    const int row = blockIdx.x;              // 0..B*M-1
    const int b   = row / M_DIM;
    const int m   = row % M_DIM;
    const int s   = posIdx[b * M_DIM + m];

    const float* x = hidden + ((size_t)b * S_DIM + (size_t)s) * H_DIM;
    const int t = threadIdx.x;

    float v[4];
    float sum = 0.f, sumsq = 0.f;
#pragma unroll
    for (int i = 0; i < 4; ++i) {
        const float f = x[t + 256 * i];
        v[i] = f;
        sum += f;
        sumsq += f * f;
    }
    // intra-wave reduction (wave32)
#pragma unroll
    for (int off = 16; off > 0; off >>= 1) {
        sum   += __shfl_xor(sum,   off, 32);
        sumsq += __shfl_xor(sumsq, off, 32);
    }
    __shared__ float s1[8], s2[8];
    const int wave = t >> 5;
    if ((t & 31) == 0) { s1[wave] = sum; s2[wave] = sumsq; }
    __syncthreads();
    sum = 0.f; sumsq = 0.f;
#pragma unroll
    for (int i = 0; i < 8; ++i) { sum += s1[i]; sumsq += s2[i]; }

    const float mu  = sum * (1.0f / H_DIM);
    const float var = sumsq * (1.0f / H_DIM) - mu * mu;
    const float inv = rsqrtf(var + LN_EPS);

    float* o = normed + (size_t)row * H_DIM;
#pragma unroll
    for (int i = 0; i < 4; ++i) {
        const int h = t + 256 * i;
        o[h] = (v[i] - mu) * inv * gamma[h] + beta[h];
    }
}

// ---------------------------------------------------------------------------
// Kernel 3: out[1024][4096] = normed[1024][1024] @ W[4096][1024]^T + bias
// using V_WMMA_F32_16X16X4_F32 with 64x64 register blocking per wave:
// a 4x4 grid of 16x16 WMMA tiles. Per K-step of 4 we load 4 A-fragments and
// 4 B-fragments (8 global_load_b64) and issue 16 WMMAs -> 128 B of L2 traffic
// per WMMA (4x less than the unblocked version).
//
// __launch_bounds__(256, 1): relax the occupancy target so the allocator can
// use the full 256 direct VGPRs; the tile needs ~180 (128 accumulator VGPRs +
// fragments + addressing). Without this the accumulators spill to scratch.
//
// A-fragment layout (16x4 f32, 2 VGPRs): lanes 0-15 hold rows M=0..15 with
// K={0,1}; lanes 16-31 the same rows with K={2,3}. B (4x16) mirrors with N
// across lanes. Both map to one contiguous float2 load per lane.
// C/D layout: lanes 0-15: N=lane, VGPR v -> M=v; lanes 16-31 -> M=v+8.
// ---------------------------------------------------------------------------
__global__ void __launch_bounds__(256, 1)
wmma_gemm_kernel(const float* __restrict__ normed,
                 const float* __restrict__ W,
                 const float* __restrict__ bias,
                 float* __restrict__ out) {
    const int wave = threadIdx.x >> 5;
    const int lane = threadIdx.x & 31;
    const int tile = blockIdx.x * 8 + wave;       // 0..1023 macro tiles
    const int rowTile = tile >> 6;                // 0..15  (64-row macro tiles)
    const int colTile = tile & 63;                // 0..63  (64-col macro tiles)

    const int mLane = lane & 15;
    const int kHalf = (lane >> 4) << 1;           // 0 or 2

    const float* Abase = normed + (size_t)(rowTile * 64 + mLane) * H_DIM + kHalf;
    const float* Bbase = W      + (size_t)(colTile * 64 + mLane) * H_DIM + kHalf;

    v8f c[4][4] = {};     // 128 accumulator VGPRs

#pragma unroll 2
    for (int k = 0; k < H_DIM; k += 4) {
        v2f a[4], b[4];
#pragma unroll
        for (int i = 0; i < 4; ++i) {
            // sub-tile strides are 16*H_DIM*4 = 64 KB -> constant instruction offsets
            a[i] = *(const v2f*)(Abase + (size_t)i * 16 * H_DIM + k);
            b[i] = *(const v2f*)(Bbase + (size_t)i * 16 * H_DIM + k);
        }
#pragma unroll
        for (int i = 0; i < 4; ++i) {
#pragma unroll
            for (int j = 0; j < 4; ++j) {
                c[i][j] = __builtin_amdgcn_wmma_f32_16x16x4_f32(
                        /*neg_a=*/false, a[i], /*neg_b=*/false, b[j],
                        /*c_mod=*/(short)0, c[i][j],
                        /*reuse_a=*/false, /*reuse_b=*/false);
            }
        }
    }

    // Epilogue: bias add + store, following the C/D VGPR layout.
    const int rSub = (lane >> 4) << 3;            // +8 for upper half lanes
#pragma unroll
    for (int j = 0; j < 4; ++j) {
        const int col = colTile * 64 + j * 16 + mLane;
        const float bv = bias[col];
#pragma unroll
        for (int i = 0; i < 4; ++i) {
            const int rbase = rowTile * 64 + i * 16 + rSub;
#pragma unroll
            for (int v = 0; v < 8; ++v) {
                out[(size_t)(rbase + v) * D_DIM + col] = c[i][j][v] + bv;
            }
        }
    }
}

// ---------------------------------------------------------------------------
// Launch. Inputs in setup_inputs() order:
//   0 hidden(f32 B,S,H) 1 gamma(f32 H) 2 beta(f32 H) 3 W(f32 D,H)
//   4 b(f32 D) 5 input_ids(int B,S) 6 mem_token_id(int scalar)
// Workspace layout: [0, 4KB) posIdx ints; [8KB, 8KB+4MB) normed panel.
// ---------------------------------------------------------------------------
extern "C" void kernel_launch(void* const* d_in, const int* in_sizes, int n_in,
                              void* d_out, int out_size, void* d_ws, size_t ws_size,
                              hipStream_t stream) {
    const float* hidden = (const float*)d_in[0];
    const float* gamma  = (const float*)d_in[1];
    const float* beta   = (const float*)d_in[2];
    const float* W      = (const float*)d_in[3];
    const float* bias   = (const float*)d_in[4];
    const int*   ids    = (const int*)d_in[5];
    const int*   memId  = (const int*)d_in[6];
    float* out = (float*)d_out;

    int*   posIdx = (int*)d_ws;
    float* normed = (float*)((char*)d_ws + 8192);

    find_mem_kernel<<<B_DIM, 32, 0, stream>>>(ids, memId, posIdx);
    ln_gather_kernel<<<B_DIM * M_DIM, 256, 0, stream>>>(hidden, gamma, beta, posIdx, normed);

    // macro tiles: (1024/64) * (4096/64) = 16 * 64 = 1024; 8 waves per block.
    wmma_gemm_kernel<<<1024 / 8, 256, 0, stream>>>(normed, W, bias, out);
}